// DWTFeature_87084756894513
// MI455X (gfx1250) — compile-verified
//
#include <hip/hip_runtime.h>

// ---------------------------------------------------------------------------
// 3-level Haar DWT (db1) as a block-diagonal 16x16 GEMM on CDNA5 WMMA:
//   D(16x16) = H16(16x16) * X(16x16),  H16 = diag(H8, H8),
//   H8 rows in output order [cA3, cD3, cD2_0, cD2_1, cD1_0..3].
// Each wave32 processes 4 tiles of 256 consecutive samples of one row using
// 4 chained V_WMMA_F32_16X16X4_F32 per tile (K = 0..15 in chunks of 4).
// D fragments are staged through LDS into flattened-wavedec order so every
// global store is a fully coalesced non-temporal b128.
// ---------------------------------------------------------------------------

typedef __attribute__((ext_vector_type(2))) float v2f;
typedef __attribute__((ext_vector_type(4))) float v4f;
typedef __attribute__((ext_vector_type(8))) float v8f;

namespace {

constexpr int kRows         = 256;
constexpr int kL            = 131072;            // samples per row
constexpr int kTileSamples  = 256;               // 16 segments x 16 samples
constexpr int kTilesPerRow  = kL / kTileSamples; // 512
constexpr int kWavesPerBlk  = 8;
constexpr int kThreads      = kWavesPerBlk * 32; // 256
constexpr int kTilesPerWave = 4;
constexpr int kTotalTiles   = kRows * kTilesPerRow;                    // 131072
constexpr int kBlocks       = kTotalTiles / (kWavesPerBlk * kTilesPerWave); // 4096
constexpr int kLdsPerWave   = kTilesPerWave * kTileSamples;            // 1024 floats

// H16[m][k], branchless: support/sign bitmasks indexed by (j*8+kk) plus two
// selects for the magnitude. No divergent control flow -> pure v_cndmask.
__device__ __forceinline__ float haar16(int m, int k) {
  const int j  = m & 7;   // output slot within 8-block
  const int kk = k & 7;   // sample within 8-block
  // byte j, bit kk (rows j=0..7 of H8): nonzero-support and negative-sign masks
  const unsigned long long SUP = 0xC0300C03F00FFFFFull;
  const unsigned long long NEG = 0x80200802C00CF000ull;
  const int bit = j * 8 + kk;
  const unsigned sup = (unsigned)(SUP >> bit) & 1u;
  const unsigned neg = (unsigned)(NEG >> bit) & 1u;
  const float C3 = 0.35355339059327373f;   // 2^-1.5  (cA3/cD3)
  const float C1 = 0.70710678118654752f;   // 2^-0.5  (cD1)
  float mag = (j < 4) ? ((j < 2) ? C3 : 0.5f) : C1;
  float v   = sup ? mag : 0.0f;
  v         = neg ? -v : v;
  // off-diagonal 8x8 blocks of H16 are zero
  return (((m ^ k) & 8) == 0) ? v : 0.0f;
}

} // namespace

__global__ __launch_bounds__(kThreads)
void dwt_haar3_wmma_kernel(const float* __restrict__ x, float* __restrict__ out) {
  __shared__ float lds[kWavesPerBlk * kLdsPerWave];   // 32 KiB / block

  const int lane = threadIdx.x & 31;
  const int wave = threadIdx.x >> 5;
  const int n    = lane & 15;  // A row / B column / D column held by this lane
  const int hi   = lane >> 4;  // lane half selects the K sub-pair (+2)

  // 4 consecutive tiles per wave; kTilesPerRow % kTilesPerWave == 0 so all
  // four tiles live in the same row. Exact grid cover -> no divergence.
  const int tile0 = (blockIdx.x * kWavesPerBlk + wave) * kTilesPerWave;
  const int row   = tile0 / kTilesPerRow;
  const int t0    = tile0 % kTilesPerRow;

  // ---- A fragments: constant per lane, hoisted out of all loops -------------
  v2f a[4];
#pragma unroll
  for (int s = 0; s < 4; ++s) {
    const int k0 = 4 * s + 2 * hi;
    a[s].x = haar16(n, k0);
    a[s].y = haar16(n, k0 + 1);
  }

  // ---- issue all 16 global b64 loads up front (read-once: non-temporal) -----
  const float* __restrict__ src =
      x + (size_t)row * kL + (size_t)t0 * kTileSamples + (size_t)n * 16;
  v2f b[kTilesPerWave][4];
#pragma unroll
  for (int u = 0; u < kTilesPerWave; ++u)
#pragma unroll
    for (int s = 0; s < 4; ++s)
      b[u][s] = __builtin_nontemporal_load(
          (const v2f*)(src + u * kTileSamples + 4 * s + 2 * hi));

  // ---- D = H16 * X : 4 chained 16x16x4 f32 WMMAs per tile -------------------
  v8f acc[kTilesPerWave];
#pragma unroll
  for (int u = 0; u < kTilesPerWave; ++u) {
    v8f c = {};
#pragma unroll
    for (int s = 0; s < 4; ++s)
      c = __builtin_amdgcn_wmma_f32_16x16x4_f32(
          /*neg_a=*/false, a[s], /*neg_b=*/false, b[u][s],
          /*c_mod=*/(short)0, c, /*reuse_a=*/false, /*reuse_b=*/false);
    acc[u] = c;
  }

  // ---- scatter D into LDS in flattened-wavedec order over the 4 tiles -------
  // C/D layout: lane holds N = lane&15, VGPR v holds M = v + 8*hi.
  // 8-block index within a tile: b2 = 2*n + hi.
  float* ls = &lds[wave * kLdsPerWave];
  const int b2 = 2 * n + hi;
#pragma unroll
  for (int u = 0; u < kTilesPerWave; ++u) {
    ls[u * 32 + b2]       = acc[u][0];                 // cA3 region [  0, 128)
    ls[128 + u * 32 + b2] = acc[u][1];                 // cD3 region [128, 256)
    v2f d2; d2.x = acc[u][2]; d2.y = acc[u][3];
    *(v2f*)(ls + 256 + u * 64 + 2 * b2) = d2;          // cD2 region [256, 512)
    v4f d1; d1.x = acc[u][4]; d1.y = acc[u][5]; d1.z = acc[u][6]; d1.w = acc[u][7];
    *(v4f*)(ls + 512 + u * 128 + 4 * b2) = d1;         // cD1 region [512,1024)
  }
  __syncthreads();

  // ---- fully coalesced non-temporal b128 stores of the contiguous runs ------
  float* __restrict__ o = out + (size_t)row * kL;
  const int l4 = 4 * lane;
  __builtin_nontemporal_store(*(const v4f*)(ls + l4),
                              (v4f*)(o + (size_t)t0 * 32 + l4));                     // cA3
  __builtin_nontemporal_store(*(const v4f*)(ls + 128 + l4),
                              (v4f*)(o + kL / 8 + (size_t)t0 * 32 + l4));            // cD3
  __builtin_nontemporal_store(*(const v4f*)(ls + 256 + l4),
                              (v4f*)(o + kL / 4 + (size_t)t0 * 64 + l4));            // cD2
  __builtin_nontemporal_store(*(const v4f*)(ls + 384 + l4),
                              (v4f*)(o + kL / 4 + (size_t)t0 * 64 + 128 + l4));
  __builtin_nontemporal_store(*(const v4f*)(ls + 512 + l4),
                              (v4f*)(o + kL / 2 + (size_t)t0 * 128 + l4));           // cD1
  __builtin_nontemporal_store(*(const v4f*)(ls + 640 + l4),
                              (v4f*)(o + kL / 2 + (size_t)t0 * 128 + 128 + l4));
  __builtin_nontemporal_store(*(const v4f*)(ls + 768 + l4),
                              (v4f*)(o + kL / 2 + (size_t)t0 * 128 + 256 + l4));
  __builtin_nontemporal_store(*(const v4f*)(ls + 896 + l4),
                              (v4f*)(o + kL / 2 + (size_t)t0 * 128 + 384 + l4));
}

extern "C" void kernel_launch(void* const* d_in, const int* in_sizes, int n_in,
                              void* d_out, int out_size, void* d_ws, size_t ws_size,
                              hipStream_t stream) {
  (void)in_sizes; (void)n_in; (void)out_size; (void)d_ws; (void)ws_size;
  const float* x = (const float*)d_in[0];
  float* out = (float*)d_out;
  dwt_haar3_wmma_kernel<<<dim3(kBlocks), dim3(kThreads), 0, stream>>>(x, out);
}